// GCNModel02_74380243632479
// MI455X (gfx1250) — compile-verified
//
#include <hip/hip_runtime.h>

#define NNODES 100000
#define NEDGES 1200000
#define FDIM   64
#define HDIM   1024
#define NGRAPH 256
#define FCAT   320   // 256 + 64

typedef float v2f __attribute__((ext_vector_type(2)));
typedef float v8f __attribute__((ext_vector_type(8)));

static inline int cdiv(long long a, long long b) { return (int)((a + b - 1) / b); }

// ---------------- degree / norm ----------------
__global__ void k_init_deg(float* deg, int n) {
    int i = blockIdx.x * blockDim.x + threadIdx.x;
    if (i < n) deg[i] = 1.0f;   // self-loop weight 1
}

__global__ void k_deg_accum(float* deg, const int* __restrict__ dst,
                            const float* __restrict__ w, int E) {
    int e = blockIdx.x * blockDim.x + threadIdx.x;
    if (e < E) atomicAdd(&deg[dst[e]], w[e]);
}

__global__ void k_norm(float* __restrict__ nrm, const float* __restrict__ deg,
                       const int* __restrict__ src, const int* __restrict__ dst,
                       const float* __restrict__ w, int E, int N) {
    int e = blockIdx.x * blockDim.x + threadIdx.x;
    if (e >= E + N) return;
    if (e < E) {
        float ds = deg[src[e]], dd = deg[dst[e]];
        nrm[e] = rsqrtf(ds) * w[e] * rsqrtf(dd);
    } else {
        nrm[e] = 1.0f / deg[e - E];   // dinv^2 * 1.0 (self loop)
    }
}

// ---------------- elementwise helpers ----------------
__global__ void k_bias_init(float* __restrict__ y, const float* __restrict__ b,
                            int total, int F) {
    int i = blockIdx.x * blockDim.x + threadIdx.x;
    if (i < total) y[i] = b[i % F];
}

__global__ void k_relu(float* __restrict__ y, int total) {
    int i = blockIdx.x * blockDim.x + threadIdx.x;
    if (i < total) y[i] = fmaxf(y[i], 0.0f);
}

__global__ void k_zero(float* __restrict__ p, int total) {
    int i = blockIdx.x * blockDim.x + threadIdx.x;
    if (i < total) p[i] = 0.0f;
}

// ---------------- fp32 WMMA GEMM:  Y[M,Nout] = X[M,K] @ W[K,Nout] (+bias,relu) ---------
// One wave computes one 16x16 output tile via V_WMMA_F32_16X16X4_F32.
// M % 16 == 0, K % 4 == 0, Nout % 16 == 0 (all shapes here satisfy this).
__global__ void k_gemm_wmma(const float* __restrict__ X, const float* __restrict__ W,
                            const float* __restrict__ bias, float* __restrict__ Y,
                            int M, int K, int Nout, int fuse_bias_relu) {
    int wavesPerBlk = blockDim.x >> 5;
    int waveId = blockIdx.x * wavesPerBlk + (threadIdx.x >> 5);
    int lane   = threadIdx.x & 31;
    int nTiles = Nout >> 4;
    int totalTiles = (M >> 4) * nTiles;
    if (waveId >= totalTiles) return;          // wave-uniform -> EXEC all-ones below
    int tn = waveId % nTiles;
    int tm = waveId / nTiles;
    int lhalf = lane >> 4;                     // 0: lanes 0-15, 1: lanes 16-31
    int l16   = lane & 15;
    int m = (tm << 4) + l16;                   // A row handled by this lane
    int n = (tn << 4) + l16;                   // B/C column handled by this lane

    const float* xrow = X + (size_t)m * K + 2 * lhalf;        // K offset base
    const float* wcol = W + (size_t)(2 * lhalf) * Nout + n;

    v8f acc = {0.f, 0.f, 0.f, 0.f, 0.f, 0.f, 0.f, 0.f};
    for (int k0 = 0; k0 < K; k0 += 4) {
        v2f a, b;
        a.x = xrow[k0];                        // X[m][k0 + 2*lhalf]
        a.y = xrow[k0 + 1];                    // X[m][k0 + 2*lhalf + 1]
        b.x = wcol[(size_t)k0 * Nout];         // W[k0 + 2*lhalf][n]
        b.y = wcol[(size_t)(k0 + 1) * Nout];   // W[k0 + 2*lhalf + 1][n]
        acc = __builtin_amdgcn_wmma_f32_16x16x4_f32(
                  false, a, false, b, (short)0, acc, false, false);
    }

    // C/D layout: VGPR r -> row tm*16 + r + 8*lhalf, column n
    float* yp = Y + (size_t)((tm << 4) + 8 * lhalf) * Nout + n;
    float bn = fuse_bias_relu ? bias[n] : 0.0f;
#pragma unroll
    for (int r = 0; r < 8; ++r) {
        float v = acc[r];
        if (fuse_bias_relu) v = fmaxf(v + bn, 0.0f);
        yp[(size_t)r * Nout] = v;
    }
}

// ---------------- edge scatter: out[dst] += h[src] * norm  (atomics, L2-resident) ------
__global__ void k_scatter(const float* __restrict__ h, const float* __restrict__ nrm,
                          const int* __restrict__ src, const int* __restrict__ dst,
                          float* __restrict__ out, int Etot, int E, int F) {
    int chunks = F >> 2;
    long long t = (long long)blockIdx.x * blockDim.x + threadIdx.x;
    if (t >= (long long)Etot * chunks) return;
    int e = (int)(t / chunks);
    int c = (int)(t % chunks);
    int s, d;
    if (e < E) { s = src[e]; d = dst[e]; }
    else       { s = d = e - E; }              // self loop
    float wgt = nrm[e];
    float4 v = *(const float4*)(h + (size_t)s * F + 4 * c);
    float* op = out + (size_t)d * F + 4 * c;
    atomicAdd(op + 0, v.x * wgt);
    atomicAdd(op + 1, v.y * wgt);
    atomicAdd(op + 2, v.z * wgt);
    atomicAdd(op + 3, v.w * wgt);
}

// ---------------- pooling: sums[b, 0:256]=x3, [256:320]=feature; counts ----------------
__global__ void k_pool(const float* __restrict__ x3, const float* __restrict__ feat,
                       const int* __restrict__ batch, float* __restrict__ sums,
                       float* __restrict__ cnts, int N) {
    const int chunks = FCAT / 4;               // 80
    long long t = (long long)blockIdx.x * blockDim.x + threadIdx.x;
    if (t >= (long long)N * chunks) return;
    int n = (int)(t / chunks);
    int c = (int)(t % chunks);
    int b = batch[n];
    int f = 4 * c;
    float4 v;
    if (f < 256) v = *(const float4*)(x3 + (size_t)n * 256 + f);
    else         v = *(const float4*)(feat + (size_t)n * FDIM + (f - 256));
    float* sp = sums + (size_t)b * FCAT + f;
    atomicAdd(sp + 0, v.x);
    atomicAdd(sp + 1, v.y);
    atomicAdd(sp + 2, v.z);
    atomicAdd(sp + 3, v.w);
    if (c == 0) atomicAdd(&cnts[b], 1.0f);
}

__global__ void k_pool_div(const float* __restrict__ sums, const float* __restrict__ cnts,
                           float* __restrict__ gc, int total) {
    int i = blockIdx.x * blockDim.x + threadIdx.x;
    if (i >= total) return;
    float cnt = fmaxf(cnts[i / FCAT], 1.0f);
    gc[i] = sums[i] / cnt;
}

// ---------------- FC2: out[b] = dot(fc1[b,:], Wf2) + bf2 ; one wave per graph ----------
__global__ void k_fc2(const float* __restrict__ a, const float* __restrict__ w,
                      const float* __restrict__ b, float* __restrict__ out, int B, int H) {
    int wave = blockIdx.x * (blockDim.x >> 5) + (threadIdx.x >> 5);
    int lane = threadIdx.x & 31;
    if (wave >= B) return;
    float s = 0.0f;
    for (int k = lane; k < H; k += 32) s += a[(size_t)wave * H + k] * w[k];
#pragma unroll
    for (int off = 16; off > 0; off >>= 1) s += __shfl_down(s, off, 32);
    if (lane == 0) out[wave] = s + b[0];
}

// ============================== host glue ==============================
extern "C" void kernel_launch(void* const* d_in, const int* in_sizes, int n_in,
                              void* d_out, int out_size, void* d_ws, size_t ws_size,
                              hipStream_t stream) {
    const float* feature = (const float*)d_in[0];     // [N,64]
    const int*   eidx    = (const int*)d_in[1];       // [2,E]
    const float* weight  = (const float*)d_in[2];     // [E]
    const int*   batch   = (const int*)d_in[3];       // [N]
    const float* W1  = (const float*)d_in[4];
    const float* b1  = (const float*)d_in[5];
    const float* W2  = (const float*)d_in[6];
    const float* b2  = (const float*)d_in[7];
    const float* W3  = (const float*)d_in[8];
    const float* b3  = (const float*)d_in[9];
    const float* Wf1 = (const float*)d_in[10];
    const float* bf1 = (const float*)d_in[11];
    const float* Wf2 = (const float*)d_in[12];
    const float* bf2 = (const float*)d_in[13];

    const int N = NNODES, E = NEDGES, Etot = NEDGES + NNODES;
    const int* src = eidx;
    const int* dst = eidx + E;

    // workspace carve-up (all 256B aligned)
    char* base = (char*)d_ws;
    size_t off = 0;
    auto alloc = [&](size_t bytes) -> float* {
        float* p = (float*)(base + off);
        off = (off + bytes + 255) & ~(size_t)255;
        return p;
    };
    float* deg  = alloc((size_t)N * 4);
    float* nrm  = alloc((size_t)Etot * 4);
    float* hbuf = alloc((size_t)N * 256 * 4);   // gemm output, reused per layer
    float* xe   = alloc((size_t)N * 256 * 4);   // x1 (64-wide) then x3 (256-wide)
    float* xo   = alloc((size_t)N * 128 * 4);   // x2 (128-wide)
    float* sums = alloc((size_t)NGRAPH * FCAT * 4);
    float* cnts = alloc((size_t)NGRAPH * 4);
    float* gc   = alloc((size_t)NGRAPH * FCAT * 4);
    float* fc1  = alloc((size_t)NGRAPH * HDIM * 4);
    (void)ws_size;

    const int T = 256;   // 8 waves per block (wave32)
    auto gemm_blocks = [&](int M, int Nout) { return cdiv((long long)(M >> 4) * (Nout >> 4), 8); };

    // 1) degrees (self-loop weight 1 folded into init)
    k_init_deg<<<cdiv(N, T), T, 0, stream>>>(deg, N);
    k_deg_accum<<<cdiv(E, T), T, 0, stream>>>(deg, dst, weight, E);
    // 2) symmetric-normalized edge weights (+ self-loop norms)
    k_norm<<<cdiv(Etot, T), T, 0, stream>>>(nrm, deg, src, dst, weight, E, N);

    // ---- layer 1: 64 -> 64 ----
    k_gemm_wmma<<<gemm_blocks(N, 64), T, 0, stream>>>(feature, W1, nullptr, hbuf, N, 64, 64, 0);
    k_bias_init<<<cdiv((long long)N * 64, T), T, 0, stream>>>(xe, b1, N * 64, 64);
    k_scatter<<<cdiv((long long)Etot * 16, T), T, 0, stream>>>(hbuf, nrm, src, dst, xe, Etot, E, 64);
    k_relu<<<cdiv((long long)N * 64, T), T, 0, stream>>>(xe, N * 64);

    // ---- layer 2: 64 -> 128 ----
    k_gemm_wmma<<<gemm_blocks(N, 128), T, 0, stream>>>(xe, W2, nullptr, hbuf, N, 64, 128, 0);
    k_bias_init<<<cdiv((long long)N * 128, T), T, 0, stream>>>(xo, b2, N * 128, 128);
    k_scatter<<<cdiv((long long)Etot * 32, T), T, 0, stream>>>(hbuf, nrm, src, dst, xo, Etot, E, 128);
    k_relu<<<cdiv((long long)N * 128, T), T, 0, stream>>>(xo, N * 128);

    // ---- layer 3: 128 -> 256 ----
    k_gemm_wmma<<<gemm_blocks(N, 256), T, 0, stream>>>(xo, W3, nullptr, hbuf, N, 128, 256, 0);
    k_bias_init<<<cdiv((long long)N * 256, T), T, 0, stream>>>(xe, b3, N * 256, 256);
    k_scatter<<<cdiv((long long)Etot * 64, T), T, 0, stream>>>(hbuf, nrm, src, dst, xe, Etot, E, 256);
    k_relu<<<cdiv((long long)N * 256, T), T, 0, stream>>>(xe, N * 256);

    // ---- global mean pool over concat([x3, feature]) ----
    k_zero<<<cdiv(NGRAPH * FCAT + NGRAPH, T), T, 0, stream>>>(sums, NGRAPH * FCAT + NGRAPH); // sums + cnts contiguous
    k_pool<<<cdiv((long long)N * (FCAT / 4), T), T, 0, stream>>>(xe, feature, batch, sums, cnts, N);
    k_pool_div<<<cdiv(NGRAPH * FCAT, T), T, 0, stream>>>(sums, cnts, gc, NGRAPH * FCAT);

    // ---- FC1 (fused bias + relu, WMMA): [256,320] @ [320,1024] ----
    k_gemm_wmma<<<gemm_blocks(NGRAPH, HDIM), T, 0, stream>>>(gc, Wf1, bf1, fc1, NGRAPH, FCAT, HDIM, 1);
    // ---- FC2: [256,1024] @ [1024,1] ----
    k_fc2<<<cdiv(NGRAPH, 8), T, 0, stream>>>(fc1, Wf2, bf2, (float*)d_out, NGRAPH, HDIM);
}